// ReconBlock_44375602102716
// MI455X (gfx1250) — compile-verified
//
#include <hip/hip_runtime.h>

#ifndef __has_builtin
#define __has_builtin(x) 0
#endif

#define T1_NB 16
#define T1_NY 256
#define T1_NX 256
#define T1_NP 3
#define T1_NQ 8
#define T1_NPIX (T1_NB * T1_NY * T1_NX)   // 1,048,576 pixels
#define T1_BLOCK 256                       // 8 wave32s per block

// gfx1250 async global->LDS path (ASYNCcnt-tracked), guarded so the file
// compiles on toolchains that lack the builtins.
#if defined(__AMDGCN__) &&                                          \
    __has_builtin(__builtin_amdgcn_global_load_async_to_lds_b128) && \
    __has_builtin(__builtin_amdgcn_s_wait_asynccnt)
#define T1_ASYNC 1
#else
#define T1_ASYNC 0
#endif

#if T1_ASYNC
// Matches the builtin's parameter type reported by hipcc:
// 'int __attribute__((__vector_size__(4*sizeof(int)))) __device__ *'
typedef int t1_v4i __attribute__((vector_size(4 * sizeof(int))));
typedef __attribute__((address_space(1))) t1_v4i t1_global_v4i;
typedef __attribute__((address_space(3))) t1_v4i t1_lds_v4i;
#endif

__global__ __launch_bounds__(T1_BLOCK) void t1_admm_xupdate_kernel(
    const float* __restrict__ x,        // [NPIX,3] (A, B, R1)
    const float* __restrict__ z,        // [NPIX,3]
    const float* __restrict__ beta,     // [NPIX,3]
    const float* __restrict__ rho_ptr,  // [1]
    const float* __restrict__ b,        // [NPIX,8]
    const float* __restrict__ tau,      // [8]
    float* __restrict__ out)            // [NPIX,3]
{
    const int tid = (int)threadIdx.x;
    const int pix = (int)blockIdx.x * T1_BLOCK + tid;  // grid sized exactly

    float4 bq0, bq1;

#if T1_ASYNC
    // Stage this block's b-tile (8 KB) into LDS asynchronously; overlap the
    // copy with the exp/Jacobian computation below. Each lane only reads back
    // its own 32-byte slot, so s_wait_asynccnt(0) is sufficient (no barrier).
    __shared__ float bsh[T1_BLOCK * T1_NQ];
    {
        float* gsrc = const_cast<float*>(b) + (size_t)pix * T1_NQ;
        float* ldst = &bsh[tid * T1_NQ];
        __builtin_amdgcn_global_load_async_to_lds_b128(
            (t1_global_v4i*)gsrc, (t1_lds_v4i*)ldst, 0, 0);
        __builtin_amdgcn_global_load_async_to_lds_b128(
            (t1_global_v4i*)(gsrc + 4), (t1_lds_v4i*)(ldst + 4), 0, 0);
    }
#else
    {
        const float4* b4 = reinterpret_cast<const float4*>(b) + (size_t)pix * 2;
        bq0 = b4[0];   // global_load_b128
        bq1 = b4[1];   // global_load_b128
    }
#endif

    // Per-pixel parameters: 12 B/lane contiguous -> global_load_b96, coalesced.
    const size_t p3 = (size_t)pix * 3;
    const float A   = x[p3 + 0];
    const float Bv  = x[p3 + 1];
    const float R1  = x[p3 + 2];
    const float z0  = z[p3 + 0],    z1  = z[p3 + 1],    z2  = z[p3 + 2];
    const float be0 = beta[p3 + 0], be1 = beta[p3 + 1], be2 = beta[p3 + 2];
    const float rho = rho_ptr[0];   // uniform -> scalar load

    float tq[T1_NQ];
#pragma unroll
    for (int q = 0; q < T1_NQ; ++q) tq[q] = tau[q];  // uniform -> scalar loads

    // ---- b-independent work (overlaps the async copy) ----
    // J_q = [1, -E_q, B*tau_q*E_q],  JTJ entries that don't need b:
    float E[T1_NQ], J2c[T1_NQ];
    float h01 = 0.f, h02 = 0.f, h11 = 0.f, h12 = 0.f, h22 = 0.f;
#pragma unroll
    for (int q = 0; q < T1_NQ; ++q) {
        const float e  = __expf(-tq[q] * R1);   // v_exp_f32
        const float j1 = -e;
        const float j2 = Bv * tq[q] * e;
        E[q]  = e;
        J2c[q] = j2;
        h01 += j1;
        h02 += j2;
        h11 += e * e;
        h12 += j1 * j2;
        h22 += j2 * j2;
    }
    // h00 = sum_q 1*1 = NQ exactly.

#if T1_ASYNC
    __builtin_amdgcn_s_wait_asynccnt(0);        // drain ASYNCcnt
    {
        const float4* l4 = reinterpret_cast<const float4*>(&bsh[tid * T1_NQ]);
        bq0 = l4[0];   // ds_load_b128
        bq1 = l4[1];   // ds_load_b128
    }
#endif

    const float bv[T1_NQ] = {bq0.x, bq0.y, bq0.z, bq0.w,
                             bq1.x, bq1.y, bq1.z, bq1.w};

    // grad1 = mean_q(J_q * r_q)
    float g0 = 0.f, g1 = 0.f, g2 = 0.f;
#pragma unroll
    for (int q = 0; q < T1_NQ; ++q) {
        const float r = (A - Bv * E[q]) - bv[q];   // s_q - b_q
        g0 += r;
        g1 -= E[q] * r;
        g2 += J2c[q] * r;
    }
    g0 = g0 * (1.f / T1_NQ) + rho * (A  + be0 - z0);
    g1 = g1 * (1.f / T1_NQ) + rho * (Bv + be1 - z1);
    g2 = g2 * (1.f / T1_NQ) + rho * (R1 + be2 - z2);

    // H = JTJ + rho*I (symmetric 3x3)
    float H00 = (float)T1_NQ + rho;
    float H11 = h11 + rho;
    float H22 = h22 + rho;
    const float H01 = h01, H02 = h02, H12 = h12;

    // Cofactors + determinant
    float c00 = H11 * H22 - H12 * H12;
    float c01 = H02 * H12 - H01 * H22;
    float c02 = H01 * H12 - H02 * H11;
    float det = H00 * c00 + H01 * c01 + H02 * c02;

    // Reference: regularize iff !(det > 0)  (covers det<=0 and NaN)
    if (!(det > 0.f)) {
        H00 += 1e-8f; H11 += 1e-8f; H22 += 1e-8f;
        c00 = H11 * H22 - H12 * H12;
        c01 = H02 * H12 - H01 * H22;
        c02 = H01 * H12 - H02 * H11;
        det = H00 * c00 + H01 * c01 + H02 * c02;
    }
    const float c11 = H00 * H22 - H02 * H02;
    const float c12 = H01 * H02 - H00 * H12;
    const float c22 = H00 * H11 - H01 * H01;
    const float inv_det = 1.f / det;

    // d = H^{-1} * grad  (adjugate / det, symmetric)
    const float d0 = (c00 * g0 + c01 * g1 + c02 * g2) * inv_det;
    const float d1 = (c01 * g0 + c11 * g1 + c12 * g2) * inv_det;
    const float d2 = (c02 * g0 + c12 * g1 + c22 * g2) * inv_det;

    out[p3 + 0] = A  - d0;
    out[p3 + 1] = Bv - d1;
    out[p3 + 2] = R1 - d2;
}

extern "C" void kernel_launch(void* const* d_in, const int* in_sizes, int n_in,
                              void* d_out, int out_size, void* d_ws, size_t ws_size,
                              hipStream_t stream) {
    // setup_inputs() order: x, z, beta, rho, sigma, b, tau
    const float* x    = (const float*)d_in[0];
    const float* z    = (const float*)d_in[1];
    const float* beta = (const float*)d_in[2];
    const float* rho  = (const float*)d_in[3];
    // d_in[4] = sigma (unused by the reference computation)
    const float* b    = (const float*)d_in[5];
    const float* tau  = (const float*)d_in[6];
    float* out = (float*)d_out;

    const int nblocks = T1_NPIX / T1_BLOCK;  // 4096, exact
    t1_admm_xupdate_kernel<<<nblocks, T1_BLOCK, 0, stream>>>(
        x, z, beta, rho, b, tau, out);
}